// NTMMemory_32676111187999
// MI455X (gfx1250) — compile-verified
//
#include <hip/hip_runtime.h>
#include <math.h>

typedef __attribute__((ext_vector_type(2))) float v2f;
typedef __attribute__((ext_vector_type(8))) float v8f;

#define OFFSET_F 1e-6f
#define COS_EPS_F 1e-8f
#define NN 8192
#define MM 64
#define BLOCK 1024
#define WAVES 32            // BLOCK / 32
#define ROWS_PER_WAVE 256   // NN / WAVES
#define TILES_PER_WAVE 16   // ROWS_PER_WAVE / 16
#define PER_THREAD 8        // NN / BLOCK

__device__ __forceinline__ float block_reduce_max(float v, float* red, int tid) {
    red[tid] = v;
    __syncthreads();
    for (int s = BLOCK / 2; s > 0; s >>= 1) {
        if (tid < s) red[tid] = fmaxf(red[tid], red[tid + s]);
        __syncthreads();
    }
    float r = red[0];
    __syncthreads();
    return r;
}

__device__ __forceinline__ float block_reduce_sum(float v, float* red, int tid) {
    red[tid] = v;
    __syncthreads();
    for (int s = BLOCK / 2; s > 0; s >>= 1) {
        if (tid < s) red[tid] += red[tid + s];
        __syncthreads();
    }
    float r = red[0];
    __syncthreads();
    return r;
}

__global__ __launch_bounds__(BLOCK) void ntm_addressing_kernel(
    const float* __restrict__ memory,   // (B, N, M)
    const float* __restrict__ k_t,      // (B, M)
    const float* __restrict__ beta_t,   // (B,)
    const float* __restrict__ g_t,      // (B,)
    const float* __restrict__ s_t,      // (B, 3)
    const float* __restrict__ gamma_t,  // (B,)
    const float* __restrict__ w_prev,   // (B, N)
    float* __restrict__ out)            // (B, N)
{
    __shared__ float X[NN];        // beta*K -> w_g buffer
    __shared__ float red[BLOCK];   // reduction scratch
    __shared__ float ksh[MM];      // k + OFFSET
    __shared__ float bnorm_sh;

    const int b   = blockIdx.x;
    const int tid = threadIdx.x;
    const int lane = tid & 31;
    const int wid  = tid >> 5;

    // ---- stage 0: key vector + its norm ----
    if (tid < MM) ksh[tid] = k_t[b * MM + tid] + OFFSET_F;
    __syncthreads();
    if (tid == 0) {
        float s = 0.f;
        for (int m = 0; m < MM; ++m) s += ksh[m] * ksh[m];
        bnorm_sh = fmaxf(sqrtf(s), COS_EPS_F);
    }
    __syncthreads();

    const float beta = beta_t[b];
    const float bne  = bnorm_sh;
    const float* memb = memory + (size_t)b * NN * MM;

    // ---- stage 1: cosine similarity via V_WMMA_F32_16X16X4_F32 ----
    // A tile: 16 rows x 4 cols of (memory + OFFSET); lane l holds row l%16,
    // K = {0,1} (lanes 0-15) or {2,3} (lanes 16-31) in VGPR0/1.
    // B: the k chunk replicated into all 16 columns, so every D column equals
    // the 16-row dot-product vector.
    const int lrow = lane & 15;
    const int koff = (lane < 16) ? 0 : 2;

    for (int t = 0; t < TILES_PER_WAVE; ++t) {
        const int row0 = wid * ROWS_PER_WAVE + t * 16;
        const float* rp = memb + (size_t)(row0 + lrow) * MM + koff;
        if (t + 1 < TILES_PER_WAVE) {
            __builtin_prefetch(rp + 16 * MM, 0, 1);   // global_prefetch_b8
        }

        v8f acc = {0.f, 0.f, 0.f, 0.f, 0.f, 0.f, 0.f, 0.f};
        float p = 0.f;  // partial sum-of-squares for row lane%16 (half the K range)
        #pragma unroll
        for (int kk = 0; kk < 16; ++kk) {
            float a0 = rp[kk * 4 + 0] + OFFSET_F;
            float a1 = rp[kk * 4 + 1] + OFFSET_F;
            p = fmaf(a0, a0, p);
            p = fmaf(a1, a1, p);
            v2f A;  A.x = a0;                    A.y = a1;
            v2f Bv; Bv.x = ksh[kk * 4 + koff];   Bv.y = ksh[kk * 4 + koff + 1];
            acc = __builtin_amdgcn_wmma_f32_16x16x4_f32(
                false, A, false, Bv, (short)0, acc, false, false);
        }

        // combine the two K-halves of the row sumsq: lane l <-> lane l^16
        float ssq = p + __shfl_xor(p, 16, 32);          // row lane%16
        // writer lane r needs row r; writer lane 16+r needs row 8+r (at lane 8+r)
        float s_row = __shfl(ssq, (lane < 16) ? lane : (lane - 8), 32);
        float inv_den = 1.0f / (fmaxf(sqrtf(s_row), COS_EPS_F) * bne);

        #pragma unroll
        for (int r = 0; r < 8; ++r) {
            if (lane == r || lane == 16 + r) {
                int row = row0 + ((lane < 16) ? r : (8 + r));
                X[row] = beta * (acc[r] * inv_den);
            }
        }
    }
    __syncthreads();

    // ---- stage 2: softmax over N (stable) ----
    float lmax = -3.402823466e38f;
    #pragma unroll
    for (int j = 0; j < PER_THREAD; ++j)
        lmax = fmaxf(lmax, X[tid + j * BLOCK]);
    const float gmax = block_reduce_max(lmax, red, tid);

    float e[PER_THREAD];
    float lsum = 0.f;
    #pragma unroll
    for (int j = 0; j < PER_THREAD; ++j) {
        e[j] = __expf(X[tid + j * BLOCK] - gmax);
        lsum += e[j];
    }
    const float Z = block_reduce_sum(lsum, red, tid);
    const float invZ = 1.0f / Z;

    // ---- stage 3: interpolate with w_{t-1}, store w_g into X ----
    const float g = g_t[b];
    #pragma unroll
    for (int j = 0; j < PER_THREAD; ++j) {
        int n = tid + j * BLOCK;
        float wc = e[j] * invZ;
        X[n] = g * wc + (1.0f - g) * w_prev[(size_t)b * NN + n];
    }
    __syncthreads();

    // ---- stage 4: 3-tap circular conv, sharpen, row sum ----
    const float s0 = s_t[b * 3 + 0];
    const float s1 = s_t[b * 3 + 1];
    const float s2 = s_t[b * 3 + 2];
    const float gam = gamma_t[b];
    float wsum_l = 0.f;
    #pragma unroll
    for (int j = 0; j < PER_THREAD; ++j) {
        int n  = tid + j * BLOCK;
        int nm = (n == 0) ? (NN - 1) : (n - 1);
        int np = (n == NN - 1) ? 0 : (n + 1);
        float wt = s0 * X[nm] + s1 * X[n] + s2 * X[np];
        wsum_l += powf(wt, gam);
    }
    const float wsum = block_reduce_sum(wsum_l, red, tid) + OFFSET_F;

    // ---- stage 5: broadcast the scalar row-sum to (b, :) ----
    #pragma unroll
    for (int j = 0; j < PER_THREAD; ++j)
        out[(size_t)b * NN + tid + j * BLOCK] = wsum;
}

extern "C" void kernel_launch(void* const* d_in, const int* in_sizes, int n_in,
                              void* d_out, int out_size, void* d_ws, size_t ws_size,
                              hipStream_t stream) {
    const float* memory  = (const float*)d_in[0];
    const float* k_t     = (const float*)d_in[1];
    const float* beta_t  = (const float*)d_in[2];
    const float* g_t     = (const float*)d_in[3];
    const float* s_t     = (const float*)d_in[4];
    const float* gamma_t = (const float*)d_in[5];
    const float* w_prev  = (const float*)d_in[6];
    float* out = (float*)d_out;

    const int B = in_sizes[2];   // beta_t has B elements
    ntm_addressing_kernel<<<B, BLOCK, 0, stream>>>(
        memory, k_t, beta_t, g_t, s_t, gamma_t, w_prev, out);
}